// fd2d_48747878810212
// MI455X (gfx1250) — compile-verified
//
#include <hip/hip_runtime.h>
#include <stdint.h>

// ---------------- problem constants (fixed by the reference) ----------------
#define NZ    26
#define NX    120
#define NBC   30
#define PADT  2
#define NZP   58           // NZ + PADT + NBC
#define NXP   180          // NX + 2*NBC
#define NCELL (NZP * NXP)  // 10440
#define NTHREADS 1024
#define CPT   11           // ceil(NCELL / NTHREADS)
#define FD_C1 (1.125f)
#define FD_C2 (-1.0f / 24.0f)

// validity / role bits per owned cell
#define MB_XF   1   // dxf valid: x in [1,177]
#define MB_XB   2   // dxb valid: x in [2,178]
#define MB_ZF   4   // dzf valid: z in [1,55]
#define MB_ZB   8   // dzb valid: z in [2,56]
#define MB_SEIS 16  // z==PADT and x in [NBC, NBC+NX)
#define MB_SRC  32  // source cell

// ---------------- gfx1250 async global->LDS path ----------------
#if defined(__has_builtin)
#if __has_builtin(__builtin_amdgcn_global_load_async_to_lds_b128)
#define USE_ASYNC_LDS 1
#endif
#endif

// builtin is strictly typed: pointer to 16-byte int vector, in global / LDS AS
typedef int v4i_ __attribute__((vector_size(4 * sizeof(int))));
typedef __attribute__((address_space(1))) v4i_ GV4;  // global v4i
typedef __attribute__((address_space(3))) v4i_ LV4;  // LDS v4i

__device__ __forceinline__ void wait_asynccnt0() {
#if defined(__has_builtin)
#if __has_builtin(__builtin_amdgcn_s_wait_asynccnt)
    __builtin_amdgcn_s_wait_asynccnt(0);
    return;
#else
    asm volatile("s_wait_asynccnt 0" ::: "memory");
    return;
#endif
#else
    asm volatile("s_wait_asynccnt 0" ::: "memory");
#endif
}

__global__ __launch_bounds__(NTHREADS, 1)
void fd2d_elastic_persistent(const int*   __restrict__ sx,
                             const int*   __restrict__ rankp,
                             const float* __restrict__ ivec,
                             const float* __restrict__ temp,
                             const float* __restrict__ swave,
                             const float* __restrict__ den,
                             const float* __restrict__ vp0,
                             const float* __restrict__ vs0,
                             const float* __restrict__ denm0,
                             const float* __restrict__ denm2,
                             float*       __restrict__ uu1,
                             float*       __restrict__ seismo,
                             float*       __restrict__ illum_out,
                             int nt)
{
    // whole wavefield state lives in LDS: 5 * 10440 * 4 B = 208,800 B (< 320 KB/WGP)
    __shared__ __attribute__((aligned(16))) float S[5 * NCELL];
    float* vx  = S;
    float* vz  = S + NCELL;
    float* txx = S + 2 * NCELL;
    float* tzz = S + 3 * NCELL;
    float* txz = S + 4 * NCELL;

    const int   tid = threadIdx.x;
    const float dtx = ivec[3];
    const int   isx = NBC + sx[rankp[0]];
    const int   src_cell = (PADT + 1) * NXP + isx;

    // ---- initial state (temp: vx,vz,txx,tzz,txz) -> LDS via async DMA loads ----
#ifdef USE_ASYNC_LDS
    {
        const int n16 = (5 * NCELL) / 4;  // 13050 x 16-byte chunks
        for (int i = tid; i < n16; i += NTHREADS) {
            __builtin_amdgcn_global_load_async_to_lds_b128(
                (GV4*)(temp + (size_t)i * 4), (LV4*)(S + (size_t)i * 4), 0, 0);
        }
    }
#else
    for (int i = tid; i < 5 * NCELL; i += NTHREADS) S[i] = temp[i];
#endif

    // ---- per-owned-cell constant coefficients into registers (overlaps DMA) ----
    float ca_r[CPT], cl_r[CPT], cm_r[CPT], b_r[CPT], dmp_r[CPT], il_r[CPT];
    int   m_r[CPT];
#pragma unroll
    for (int k = 0; k < CPT; ++k) {
        const int c = tid + k * NTHREADS;
        ca_r[k] = cl_r[k] = cm_r[k] = b_r[k] = dmp_r[k] = 0.0f;
        il_r[k] = 0.0f;
        m_r[k]  = 0;
        if (c < NCELL) {
            const int z = c / NXP;
            const int x = c - z * NXP;
            // replicate-pad mapping into the 26x120 model
            int zi = z - PADT; zi = zi < 0 ? 0 : (zi > NZ - 1 ? NZ - 1 : zi);
            int xi = x - NBC;  xi = xi < 0 ? 0 : (xi > NX - 1 ? NX - 1 : xi);
            const float vp = vp0[zi * NX + xi];
            const float vs = vs0[zi * NX + xi];
            const float dn = den[c];
            float ca = vp * vp * dn;
            float cm = vs * vs * dn;
            const float cl = (ca - 2.0f * cm) * denm0[c];   // cl uses pre-patch ca
            if (z == PADT - 1) ca = 2.0f * cm;              // free-surface patch
            const float b = dtx * (1.0f / dn) * denm2[c];
            // Cerjan sponge
            int dzb = z - (NZP - NBC - 1); if (dzb < 0) dzb = 0;
            int dxl = NBC - x;             if (dxl < 0) dxl = 0;
            int dxr = x - (NXP - NBC - 1); if (dxr < 0) dxr = 0;
            int dd = dzb > dxl ? dzb : dxl; if (dxr > dd) dd = dxr;
            const float e = 0.015f * (float)dd;
            const float damp = expf(-e * e);

            ca_r[k] = ca; cl_r[k] = cl; cm_r[k] = cm; b_r[k] = b; dmp_r[k] = damp;

            int m = 0;
            if (x >= 1 && x <= NXP - 3) m |= MB_XF;
            if (x >= 2 && x <= NXP - 2) m |= MB_XB;
            if (z >= 1 && z <= NZP - 3) m |= MB_ZF;
            if (z >= 2 && z <= NZP - 2) m |= MB_ZB;
            if (z == PADT && x >= NBC && x < NBC + NX) m |= MB_SEIS;
            if (c == src_cell) m |= MB_SRC;
            m_r[k] = m;
        }
    }

#ifdef USE_ASYNC_LDS
    wait_asynccnt0();
#endif
    __syncthreads();

    // ---------------- nt sequential timesteps, 2 barriers each ----------------
    for (int t = 0; t < nt; ++t) {
        const float srcv = swave[t];

        // phase 1: velocity update (reads stress neighbors, writes own velocity)
#pragma unroll
        for (int k = 0; k < CPT; ++k) {
            const int c = tid + k * NTHREADS;
            if (c < NCELL) {
                const int m = m_r[k];
                const float dxb_txx = (m & MB_XB)
                    ? FD_C1 * (txx[c] - txx[c - 1]) + FD_C2 * (txx[c + 1] - txx[c - 2]) : 0.0f;
                const float dzb_txz = (m & MB_ZB)
                    ? FD_C1 * (txz[c] - txz[c - NXP]) + FD_C2 * (txz[c + NXP] - txz[c - 2 * NXP]) : 0.0f;
                const float dzf_tzz = (m & MB_ZF)
                    ? FD_C1 * (tzz[c + NXP] - tzz[c]) + FD_C2 * (tzz[c + 2 * NXP] - tzz[c - NXP]) : 0.0f;
                const float dxb_txz = (m & MB_XB)
                    ? FD_C1 * (txz[c] - txz[c - 1]) + FD_C2 * (txz[c + 1] - txz[c - 2]) : 0.0f;

                const float vxn = (vx[c] + b_r[k] * (dxb_txx + dzb_txz)) * dmp_r[k];
                const float vzn = (vz[c] + b_r[k] * (dzf_tzz + dxb_txz)) * dmp_r[k];
                vx[c] = vxn;
                vz[c] = vzn;
                uu1[(size_t)t * NCELL + c] = vzn;                    // streamed output
                if (m & MB_SEIS)
                    seismo[(size_t)t * NX + (c - (PADT * NXP + NBC))] = vzn;
            }
        }
        __syncthreads();

        // phase 2: stress update (reads velocity neighbors, writes own stress)
#pragma unroll
        for (int k = 0; k < CPT; ++k) {
            const int c = tid + k * NTHREADS;
            if (c < NCELL) {
                const int m = m_r[k];
                const float dvxdx = (m & MB_XB)
                    ? FD_C1 * (vx[c] - vx[c - 1]) + FD_C2 * (vx[c + 1] - vx[c - 2]) : 0.0f;
                const float dvzdz = (m & MB_ZB)
                    ? FD_C1 * (vz[c] - vz[c - NXP]) + FD_C2 * (vz[c + NXP] - vz[c - 2 * NXP]) : 0.0f;
                const float dzf_vx = (m & MB_ZF)
                    ? FD_C1 * (vx[c + NXP] - vx[c]) + FD_C2 * (vx[c + 2 * NXP] - vx[c - NXP]) : 0.0f;
                const float dxf_vz = (m & MB_XF)
                    ? FD_C1 * (vz[c + 1] - vz[c]) + FD_C2 * (vz[c + 2] - vz[c - 1]) : 0.0f;

                float txxn = (txx[c] + dtx * (ca_r[k] * dvxdx + cl_r[k] * dvzdz)) * dmp_r[k];
                float tzzn = (tzz[c] + dtx * (cl_r[k] * dvxdx + ca_r[k] * dvzdz)) * dmp_r[k];
                const float txzn = (txz[c] + dtx * cm_r[k] * (dzf_vx + dxf_vz)) * dmp_r[k];
                if (m & MB_SRC) { txxn += srcv; tzzn += srcv; }     // after damp, ref order
                txx[c] = txxn;
                tzz[c] = tzzn;
                txz[c] = txzn;

                const float dv = dvxdx + dvzdz;
                il_r[k] += dv * dv;                                  // illumination
            }
        }
        __syncthreads();
    }

    // ---- final illumination write ----
#pragma unroll
    for (int k = 0; k < CPT; ++k) {
        const int c = tid + k * NTHREADS;
        if (c < NCELL) illum_out[c] = il_r[k];
    }
}

// ---------------------------------------------------------------------------
extern "C" void kernel_launch(void* const* d_in, const int* in_sizes, int n_in,
                              void* d_out, int out_size, void* d_ws, size_t ws_size,
                              hipStream_t stream) {
    // setup_inputs order: sx, rank, input_vector, temp, s, den, device, vp0, vs0, denm0, denm2
    const int*   sx    = (const int*)d_in[0];
    const int*   rankp = (const int*)d_in[1];
    const float* ivec  = (const float*)d_in[2];
    const float* temp  = (const float*)d_in[3];
    const float* s     = (const float*)d_in[4];
    const float* den   = (const float*)d_in[5];
    const float* vp0   = (const float*)d_in[7];
    const float* vs0   = (const float*)d_in[8];
    const float* denm0 = (const float*)d_in[9];
    const float* denm2 = (const float*)d_in[10];

    const int nt = in_sizes[4];  // length of Ricker wavelet == NT

    // outputs concatenated flat: uu1 (nt,58,180), seismo (nt,120), illum (58,180)
    float* uu1    = (float*)d_out;
    float* seismo = uu1 + (size_t)nt * NCELL;
    float* illum  = seismo + (size_t)nt * NX;

    hipLaunchKernelGGL(fd2d_elastic_persistent, dim3(1), dim3(NTHREADS), 0, stream,
                       sx, rankp, ivec, temp, s, den, vp0, vs0, denm0, denm2,
                       uu1, seismo, illum, nt);
}